// GCNv1_1571958030450
// MI455X (gfx1250) — compile-verified
//
#include <hip/hip_runtime.h>
#include <hip/hip_bf16.h>
#include <math.h>

typedef float v2f __attribute__((ext_vector_type(2)));
typedef float v8f __attribute__((ext_vector_type(8)));

#define N_NODES 100000
#define N_EDGES 1600000
#define IN_DIM  128
#define HID_DIM 64
#define OUT_DIM 40
#define OUT_PAD 48   // 3 tiles of 16, cols 40..47 zero-padded

// ---------------------------------------------------------------- degree ----
__global__ void init_deg_kernel(float* deg, int n) {
    int i = blockIdx.x * blockDim.x + threadIdx.x;
    if (i < n) deg[i] = 1.0f;            // self-loop contributes 1
}

__global__ void count_deg_kernel(const int* __restrict__ dst, float* __restrict__ deg, int E) {
    int i = blockIdx.x * blockDim.x + threadIdx.x;
    if (i < E) unsafeAtomicAdd(&deg[dst[i]], 1.0f);
}

__global__ void rsqrt_deg_kernel(float* deg, int n) {
    int i = blockIdx.x * blockDim.x + threadIdx.x;
    if (i < n) deg[i] = rsqrtf(deg[i]);  // deg >= 1 always (self-loop)
}

__global__ void zero_kernel(float* p, int n) {
    int i = blockIdx.x * blockDim.x + threadIdx.x;
    if (i < n) p[i] = 0.0f;
}

// ------------------------------------------------------------ WMMA GEMM ----
// Hs[M x N] = (X[M x in_stride(:,0..K)] @ W[K x Nsrc, zero-padded to N]) * dis[row]
//
// One wave computes a 16-row strip across all NT 16-col tiles.
// A 16x4 f32 frag: lanes 0-15 rows, vgpr{0,1} = K {k0,k0+1} (lanes<16) /
//                  {k0+2,k0+3} (lanes>=16)  -> one aligned float2 global load.
// B 4x16 f32 frag: stored in LDS pre-swizzled fragment-major: for each
//                  (kk,nt) tile, 64 floats in [lane][2] order so each lane's
//                  operand pair is ONE ds_load_b64 at base + lane*8 with an
//                  immediate tile offset (conflict-free: 8B lane stride hits
//                  all 64 banks exactly once).
// C/D 16x16 f32: vgpr r = row r (lanes<16) / row r+8 (lanes>=16), col = lane&15.
template <int K, int NT>
__global__ void gemm_wmma_scale_kernel(const float* __restrict__ X,
                                       const float* __restrict__ W,
                                       const float* __restrict__ dis,
                                       float* __restrict__ Hs,
                                       int M, int in_stride, int Nsrc) {
    constexpr int N = NT * 16;
    constexpr int KSTEPS = K / 4;
    constexpr int FRAG = 64;                     // floats per (kk,nt) fragment
    __shared__ float ldsWf[KSTEPS * NT * FRAG];  // == K*N floats

    // Stage W into LDS in fragment-major order, zero-padding cols >= Nsrc.
    for (int i = threadIdx.x; i < KSTEPS * NT * FRAG; i += blockDim.x) {
        int kk   = i / (NT * FRAG);
        int rem  = i - kk * (NT * FRAG);
        int nt   = rem / FRAG;
        int p    = rem - nt * FRAG;
        int ln   = p >> 1;                 // lane 0..31
        int j    = p & 1;                  // vgpr pair element
        int koff = (ln >> 4) << 1;         // 0 or 2
        int wrow = kk * 4 + koff + j;
        int wcol = nt * 16 + (ln & 15);
        ldsWf[i] = (wcol < Nsrc) ? W[wrow * Nsrc + wcol] : 0.0f;
    }
    __syncthreads();

    const int wave = threadIdx.x >> 5;
    const int lane = threadIdx.x & 31;
    const int rowbase = (blockIdx.x * 8 + wave) * 16;
    if (rowbase >= M) return;            // whole-wave uniform exit

    const int mlane = lane & 15;
    const int koff  = (lane >> 4) << 1;  // 0 or 2
    int row = rowbase + mlane;
    if (row >= M) row = M - 1;           // clamp: keep EXEC all-1s for WMMA
    const float* xrow = X + (long long)row * in_stride;
    __builtin_prefetch(xrow, 0, 1);      // global_prefetch_b8

    const float* wlane = ldsWf + (lane << 1);   // per-lane LDS base (b64-aligned)

    v8f acc[NT] = {};

#pragma unroll
    for (int kk = 0; kk < KSTEPS; ++kk) {
        v2f a = *(const v2f*)(xrow + kk * 4 + koff);          // 8B-aligned b64
#pragma unroll
        for (int nt = 0; nt < NT; ++nt) {
            v2f b = *(const v2f*)(wlane + (kk * NT + nt) * FRAG);  // ds_load_b64
            acc[nt] = __builtin_amdgcn_wmma_f32_16x16x4_f32(
                false, a, false, b, (short)0, acc[nt], false, false);
        }
    }

    // Epilogue: scale rows by dis[] and store (guard only the tail rows).
#pragma unroll
    for (int nt = 0; nt < NT; ++nt) {
#pragma unroll
        for (int r = 0; r < 8; ++r) {
            int m = (lane < 16) ? r : (r + 8);
            int grow = rowbase + m;
            if (grow < M)
                Hs[(long long)grow * N + nt * 16 + mlane] = acc[nt][r] * dis[grow];
        }
    }
}

// -------------------------------------------------------- edge scatter -----
// acc[dst,:] += Hs[src,:]; one thread per (edge, 4 features): float4 gather,
// 4x global_atomic_add_f32.
__global__ void scatter_add_kernel(const int* __restrict__ src,
                                   const int* __restrict__ dst,
                                   const float* __restrict__ Hs,
                                   float* __restrict__ acc,
                                   int E, int F) {
    int t = blockIdx.x * blockDim.x + threadIdx.x;
    int per = F >> 2;
    if (t >= E * per) return;
    int e = t / per;
    int q = (t - e * per) << 2;
    int s = src[e], d = dst[e];
    float4 v = *(const float4*)(Hs + (long long)s * F + q);
    float* ap = acc + (long long)d * F + q;
    unsafeAtomicAdd(ap + 0, v.x);
    unsafeAtomicAdd(ap + 1, v.y);
    unsafeAtomicAdd(ap + 2, v.z);
    unsafeAtomicAdd(ap + 3, v.w);
}

// ------------------------------------------------- layer-1 finalize/ReLU ---
// Hs <- relu(dis[row]*(acc + Hs) + b[col])   (in place; Hs row is the self-loop term)
__global__ void finalize_relu_kernel(float* __restrict__ Hs,
                                     const float* __restrict__ acc,
                                     const float* __restrict__ dis,
                                     const float* __restrict__ b,
                                     int M, int F) {
    int t = blockIdx.x * blockDim.x + threadIdx.x;
    if (t >= M * F) return;
    int row = t / F;
    int f = t - row * F;
    float v = dis[row] * (acc[t] + Hs[t]) + b[f];
    Hs[t] = fmaxf(v, 0.0f);
}

// --------------------------------- layer-2 finalize + log_softmax (40) -----
__global__ void finalize_logsoftmax_kernel(const float* __restrict__ Hs2,  // stride OUT_PAD
                                           const float* __restrict__ acc2, // stride OUT_PAD
                                           const float* __restrict__ dis,
                                           const float* __restrict__ b,
                                           float* __restrict__ out, int M) {
    int i = blockIdx.x * blockDim.x + threadIdx.x;
    if (i >= M) return;
    float di = dis[i];
    float vals[OUT_DIM];
    float mx = -INFINITY;
    long long base = (long long)i * OUT_PAD;
#pragma unroll
    for (int f = 0; f < OUT_DIM; ++f) {
        float v = di * (acc2[base + f] + Hs2[base + f]) + b[f];
        vals[f] = v;
        mx = fmaxf(mx, v);
    }
    float s = 0.0f;
#pragma unroll
    for (int f = 0; f < OUT_DIM; ++f) s += __expf(vals[f] - mx);
    float lse = mx + __logf(s);
    long long obase = (long long)i * OUT_DIM;
#pragma unroll
    for (int f = 0; f < OUT_DIM; ++f) out[obase + f] = vals[f] - lse;
}

// ---------------------------------------------------------------- launch ---
extern "C" void kernel_launch(void* const* d_in, const int* in_sizes, int n_in,
                              void* d_out, int out_size, void* d_ws, size_t ws_size,
                              hipStream_t stream) {
    const float* x   = (const float*)d_in[0];
    const int*   ei  = (const int*)d_in[1];   // [2][E]: row 0 = src, row 1 = dst
    const float* W1  = (const float*)d_in[2];
    const float* b1  = (const float*)d_in[3];
    const float* W2  = (const float*)d_in[4];
    const float* b2  = (const float*)d_in[5];
    float* out = (float*)d_out;

    const int* srcp = ei;
    const int* dstp = ei + N_EDGES;

    // Workspace carve-up (floats): dis | Hs1 (N*64) | bufA (N*64) | acc2 (N*48)
    float* ws   = (float*)d_ws;
    float* dis  = ws;                          // N_NODES (padded to 102400)
    float* Hs1  = dis + 102400;                // N*64: Hs1, then relu(out1) in place
    float* bufA = Hs1 + (size_t)N_NODES * 64;  // N*64: acc1, then reused as Hs2 (stride 48)
    float* acc2 = bufA + (size_t)N_NODES * 64; // N*48

    const int T = 256;
    dim3 blk(T);

    // degrees -> dis = rsqrt(deg)
    init_deg_kernel<<<(N_NODES + T - 1) / T, blk, 0, stream>>>(dis, N_NODES);
    count_deg_kernel<<<(N_EDGES + T - 1) / T, blk, 0, stream>>>(dstp, dis, N_EDGES);
    rsqrt_deg_kernel<<<(N_NODES + T - 1) / T, blk, 0, stream>>>(dis, N_NODES);

    const int rowBlocks = (N_NODES + 8 * 16 - 1) / (8 * 16);   // 8 waves/block, 16 rows/wave

    // ---- layer 1 ----
    gemm_wmma_scale_kernel<IN_DIM, 4><<<rowBlocks, blk, 0, stream>>>(
        x, W1, dis, Hs1, N_NODES, IN_DIM, HID_DIM);
    zero_kernel<<<(N_NODES * 64 + T - 1) / T, blk, 0, stream>>>(bufA, N_NODES * 64);
    scatter_add_kernel<<<(N_EDGES * 16 + T - 1) / T, blk, 0, stream>>>(
        srcp, dstp, Hs1, bufA, N_EDGES, 64);
    finalize_relu_kernel<<<(N_NODES * 64 + T - 1) / T, blk, 0, stream>>>(
        Hs1, bufA, dis, b1, N_NODES, 64);      // Hs1 now holds relu(out1)

    // ---- layer 2 ----  (bufA reused as Hs2 with stride OUT_PAD=48)
    gemm_wmma_scale_kernel<HID_DIM, 3><<<rowBlocks, blk, 0, stream>>>(
        Hs1, W2, dis, bufA, N_NODES, HID_DIM, OUT_DIM);
    zero_kernel<<<(N_NODES * OUT_PAD + T - 1) / T, blk, 0, stream>>>(acc2, N_NODES * OUT_PAD);
    scatter_add_kernel<<<(N_EDGES * (OUT_PAD / 4) + T - 1) / T, blk, 0, stream>>>(
        srcp, dstp, bufA, acc2, N_EDGES, OUT_PAD);
    finalize_logsoftmax_kernel<<<(N_NODES + T - 1) / T, blk, 0, stream>>>(
        bufA, acc2, dis, b2, out, N_NODES);
}